// CorticalMap_65584150610094
// MI455X (gfx1250) — compile-verified
//
#include <hip/hip_runtime.h>
#include <hip/hip_bf16.h>

// ---------------- problem constants ----------------
#define S      128
#define AK     21
#define AC     2
#define LK     41
#define EX     9
#define NITERS 5
#define MAXR   19
#define PP     882        // AC*AK*AK
#define KK     1681       // LK*LK
#define HH     148        // S + AK - 1

typedef float vf2 __attribute__((ext_vector_type(2)));
typedef float vf8 __attribute__((ext_vector_type(8)));

__device__ __forceinline__ int refl(int i) {
    if (i < 0)  return -i;
    if (i >= S) return 2 * S - 2 - i;
    return i;
}

// =====================================================================
// Affinity kernel: per-pixel dot of unfolded x (times envelope) with that
// pixel's private RF row via V_WMMA_F32_16X16X4_F32 (diag extraction),
// plus a second WMMA with A=ones giving per-pixel raw-weight row sums
// (rfs normalization folded in). Writes x_tiles with non-temporal stores
// (write-once data; keep it out of L2 so lat_weights can stay resident).
// rfs is read-once -> non-temporal loads.
// Tile: 16 wide x 8 tall; 8 waves, one wave per pixel row.
// =====================================================================
__global__ __launch_bounds__(256)
void cm_aff_kernel(const float* __restrict__ x,      // (AC,HH,HH)
                   const float* __restrict__ rfs,    // (S*S, PP) raw
                   const float* __restrict__ envA,   // (PP)
                   const float* __restrict__ adath,  // (S*S)
                   float* __restrict__ raw_out,      // (S*S)
                   float* __restrict__ aff_out,      // (S*S)  ws
                   float* __restrict__ lat_out,      // (S*S)  ws
                   float* __restrict__ xt_out)       // (S*S, PP)
{
    const int j0   = blockIdx.x * 16;
    const int i0   = blockIdx.y * 8;
    const int tid  = threadIdx.x;
    const int wave = tid >> 5;
    const int lane = tid & 31;
    const int m    = lane & 15;
    const int hi   = lane >> 4;

    __shared__ float win[AC * 28 * 36];   // x window: rows i0..i0+27, cols j0..j0+35
    __shared__ float env[PP];
    __shared__ int   off[PP];

    for (int idx = tid; idx < AC * 28 * 36; idx += 256) {
        int c   = idx / (28 * 36);
        int rem = idx % (28 * 36);
        int wy  = rem / 36, wx = rem % 36;
        win[idx] = x[c * HH * HH + (i0 + wy) * HH + (j0 + wx)];
    }
    for (int k = tid; k < PP; k += 256) {
        env[k] = envA[k];
        int c = k / (AK * AK);
        int r = (k % (AK * AK)) / AK;
        int v = k % AK;
        off[k] = c * 28 * 36 + r * 36 + v;
    }
    __syncthreads();

    const int li = i0 + wave;
    const int lm = li * S + j0 + m;               // pixel for both A-row m and B-col m
    const float* __restrict__ Wrow = rfs + (size_t)lm * PP;
    float* __restrict__ Xrow = xt_out + (size_t)lm * PP;
    const int abase = wave * 36 + m;

    vf8 acc  = {0.f,0.f,0.f,0.f,0.f,0.f,0.f,0.f};
    vf8 accS = {0.f,0.f,0.f,0.f,0.f,0.f,0.f,0.f};
    const vf2 ones = {1.f, 1.f};

    const int KMAIN = (PP / 4) * 4;               // 880
    for (int cb = 0; cb < KMAIN; cb += 4) {
        __builtin_prefetch(Wrow + cb + 96, 0, 0);
        // A layout (doc table): lanes0-15 VGPR{0,1}=K{cb,cb+1}; lanes16-31 = K{cb+2,cb+3}
        int ka = cb + 2 * hi;
        float ax = win[abase + off[ka]]     * env[ka];
        float ay = win[abase + off[ka + 1]] * env[ka + 1];
        // x_tiles = unfold(x)*envelope == the A values (write-once -> NT store)
        vf2 xtv = {ax, ay};
        __builtin_nontemporal_store(xtv, (vf2*)(Xrow + ka));
        vf2 a = {ax, ay};
        // B layout: VGPR0 lanes0-15 K=cb / lanes16-31 K=cb+1 ; VGPR1 K=cb+2 / cb+3 ; N=lane&15
        vf2 b = {__builtin_nontemporal_load(Wrow + cb + hi),
                 __builtin_nontemporal_load(Wrow + cb + 2 + hi)};
        acc  = __builtin_amdgcn_wmma_f32_16x16x4_f32(false, a,    false, b, (short)0, acc,  false, false);
        accS = __builtin_amdgcn_wmma_f32_16x16x4_f32(false, ones, false, b, (short)0, accS, false, false);
    }
    { // remainder chunk: valid k = 880, 881
        int cb = KMAIN;
        int ka = cb + 2 * hi;
        vf2 a = {0.f, 0.f};
        vf2 b = {0.f, 0.f};
        if (ka < PP) {
            float ax = win[abase + off[ka]] * env[ka];
            float ay = (ka + 1 < PP) ? win[abase + off[ka + 1]] * env[ka + 1] : 0.f;
            a.x = ax; a.y = ay;
            if (ka + 1 < PP) {
                vf2 xtv = {ax, ay};
                __builtin_nontemporal_store(xtv, (vf2*)(Xrow + ka));
            } else {
                __builtin_nontemporal_store(ax, Xrow + ka);
            }
        }
        if (cb + hi     < PP) b.x = __builtin_nontemporal_load(Wrow + cb + hi);
        if (cb + 2 + hi < PP) b.y = __builtin_nontemporal_load(Wrow + cb + 2 + hi);
        acc  = __builtin_amdgcn_wmma_f32_16x16x4_f32(false, a,    false, b, (short)0, acc,  false, false);
        accS = __builtin_amdgcn_wmma_f32_16x16x4_f32(false, ones, false, b, (short)0, accS, false, false);
    }

    // diagonal extraction: D[m,m] -> lane m (m<8: VGPR m lane m; m>=8: VGPR m-8 lane m+16)
    float dotv = 0.f, sumv = 0.f;
#pragma unroll
    for (int r = 0; r < 8; ++r) {
        float dlo = __shfl(acc[r],  r,      32);
        float dhi = __shfl(acc[r],  r + 24, 32);
        float slo = __shfl(accS[r], r,      32);
        float shi = __shfl(accS[r], r + 24, 32);
        if (lane == r)     { dotv = dlo; sumv = slo; }
        if (lane == r + 8) { dotv = dhi; sumv = shi; }
    }
    if (lane < 16) {
        int l = li * S + j0 + lane;
        float raw  = dotv * (float)PP / sumv;     // dot with normalized rfs
        float affv = dotv / sumv - adath[l];      // raw/P - adathresh
        raw_out[l] = raw;
        aff_out[l] = affv;
        lat_out[l] = fmaxf(affv, 0.f);
    }
}

// =====================================================================
// 9x9 shared-kernel excitatory conv with reflect padding (pad=4).
// Tile 16x16, window 24x24 in LDS.
// =====================================================================
__global__ __launch_bounds__(256)
void cm_sre_kernel(const float* __restrict__ latIn,
                   const float* __restrict__ sre,   // (81)
                   float* __restrict__ latOut)
{
    const int j0 = blockIdx.x * 16, i0 = blockIdx.y * 16;
    const int tid = threadIdx.x;
    const int tx = tid & 15, ty = tid >> 4;

    __shared__ float win[24 * 24];
    __shared__ float kf[EX * EX];
    for (int idx = tid; idx < 24 * 24; idx += 256) {
        int wy = idx / 24, wx = idx % 24;
        win[idx] = latIn[refl(i0 + wy - 4) * S + refl(j0 + wx - 4)];
    }
    if (tid < EX * EX) kf[tid] = sre[tid];
    __syncthreads();

    float s = 0.f;
#pragma unroll
    for (int u = 0; u < EX; ++u)
#pragma unroll
        for (int v = 0; v < EX; ++v)
            s = fmaf(win[(ty + u) * 24 + tx + v], kf[u * EX + v], s);
    latOut[(i0 + ty) * S + j0 + tx] = s;
}

// =====================================================================
// Lateral inhibition pass (the big stream: 110MB of lat_weights, re-read
// every iteration -> default RT policy so it becomes/stays L2-resident).
// Per-pixel 1681-weight dot via WMMA diag + row-sum WMMA (normalization
// folded in: lat_neg = dot / rowsum). Fused: out = relu(latE - neg + aff)*1.5
// Tile 16 wide x 8 tall; zero-padded latE window 48x56 in LDS.
// =====================================================================
__global__ __launch_bounds__(256)
void cm_lat_kernel(const float* __restrict__ latE,
                   const float* __restrict__ Wlat,   // (S*S, KK) raw
                   const float* __restrict__ envK,   // (KK)
                   const float* __restrict__ aff,
                   float* __restrict__ latOut)
{
    const int j0   = blockIdx.x * 16;
    const int i0   = blockIdx.y * 8;
    const int tid  = threadIdx.x;
    const int wave = tid >> 5;
    const int lane = tid & 31;
    const int m    = lane & 15;
    const int hi   = lane >> 4;

    __shared__ float win[48 * 56];   // rows i0-20..i0+27, cols j0-20..j0+35, zero pad
    __shared__ float env[KK];
    __shared__ int   off[KK];

    for (int idx = tid; idx < 48 * 56; idx += 256) {
        int wy = idx / 56, wx = idx % 56;
        int gi = i0 + wy - 20, gj = j0 + wx - 20;
        float v = 0.f;
        if (gi >= 0 && gi < S && gj >= 0 && gj < S) v = latE[gi * S + gj];
        win[idx] = v;
    }
    for (int k = tid; k < KK; k += 256) {
        env[k] = envK[k];
        off[k] = (k / LK) * 56 + (k % LK);
    }
    __syncthreads();

    const int li = i0 + wave;
    const int lm = li * S + j0 + m;
    const float* __restrict__ Wrow = Wlat + (size_t)lm * KK;
    const int abase = wave * 56 + m;

    vf8 acc  = {0.f,0.f,0.f,0.f,0.f,0.f,0.f,0.f};
    vf8 accS = {0.f,0.f,0.f,0.f,0.f,0.f,0.f,0.f};
    const vf2 ones = {1.f, 1.f};

    const int KMAIN = (KK / 4) * 4;   // 1680
    for (int cb = 0; cb < KMAIN; cb += 4) {
        __builtin_prefetch(Wrow + cb + 96, 0, 0);
        int ka = cb + 2 * hi;
        vf2 a = {win[abase + off[ka]]     * env[ka],
                 win[abase + off[ka + 1]] * env[ka + 1]};
        vf2 b = {Wrow[cb + hi], Wrow[cb + 2 + hi]};
        acc  = __builtin_amdgcn_wmma_f32_16x16x4_f32(false, a,    false, b, (short)0, acc,  false, false);
        accS = __builtin_amdgcn_wmma_f32_16x16x4_f32(false, ones, false, b, (short)0, accS, false, false);
    }
    { // remainder: only k = 1680 valid
        int cb = KMAIN;
        int ka = cb + 2 * hi;
        vf2 a = {0.f, 0.f};
        vf2 b = {0.f, 0.f};
        if (ka < KK)     a.x = win[abase + off[ka]]     * env[ka];
        if (ka + 1 < KK) a.y = win[abase + off[ka + 1]] * env[ka + 1];
        if (cb + hi     < KK) b.x = Wrow[cb + hi];
        if (cb + 2 + hi < KK) b.y = Wrow[cb + 2 + hi];
        acc  = __builtin_amdgcn_wmma_f32_16x16x4_f32(false, a,    false, b, (short)0, acc,  false, false);
        accS = __builtin_amdgcn_wmma_f32_16x16x4_f32(false, ones, false, b, (short)0, accS, false, false);
    }

    float dotv = 0.f, sumv = 0.f;
#pragma unroll
    for (int r = 0; r < 8; ++r) {
        float dlo = __shfl(acc[r],  r,      32);
        float dhi = __shfl(acc[r],  r + 24, 32);
        float slo = __shfl(accS[r], r,      32);
        float shi = __shfl(accS[r], r + 24, 32);
        if (lane == r)     { dotv = dlo; sumv = slo; }
        if (lane == r + 8) { dotv = dhi; sumv = shi; }
    }
    if (lane < 16) {
        int l = li * S + j0 + lane;
        float lat_neg = dotv / sumv;                       // tiles . (raw/rowsum)
        float center  = win[(wave + 20) * 56 + lane + 20]; // latE[l]
        float v = center - lat_neg + aff[l];
        latOut[l] = fmaxf(v, 0.f) * 1.5f;
    }
}

// =====================================================================
// Masked 19x19 max normalization (reflect pad 9) + running mean.
// Tile 16x16, window 34x34 in LDS.
// =====================================================================
__global__ __launch_bounds__(256)
void cm_max_kernel(const float* __restrict__ latIn,
                   const float* __restrict__ mmask,   // (361)
                   float* __restrict__ latNext,       // ws: normalized lat
                   float* __restrict__ meanBuf,       // ws: sum over iters
                   float* __restrict__ latFinal,      // d_out+16384
                   int first)
{
    const int j0 = blockIdx.x * 16, i0 = blockIdx.y * 16;
    const int tid = threadIdx.x;
    const int tx = tid & 15, ty = tid >> 4;

    __shared__ float win[34 * 34];
    __shared__ float mk[MAXR * MAXR];
    for (int idx = tid; idx < 34 * 34; idx += 256) {
        int wy = idx / 34, wx = idx % 34;
        win[idx] = latIn[refl(i0 + wy - 9) * S + refl(j0 + wx - 9)];
    }
    for (int idx = tid; idx < MAXR * MAXR; idx += 256) mk[idx] = mmask[idx];
    __syncthreads();

    float mx = 0.f;  // all values >= 0 (post-relu*1.5)
    for (int u = 0; u < MAXR; ++u)
        for (int v = 0; v < MAXR; ++v)
            mx = fmaxf(mx, win[(ty + u) * 34 + tx + v] * mk[u * MAXR + v]);
    float lat_max = fmaxf(mx, 1.f);
    float center  = win[(ty + 9) * 34 + tx + 9];
    float latn    = center / (lat_max + 1e-5f);

    int l = (i0 + ty) * S + j0 + tx;
    float mean = first ? latn : (meanBuf[l] + latn);
    meanBuf[l]  = mean;
    latNext[l]  = latn;
    latFinal[l] = latn;   // last iteration's value is the "lat" output
}

// =====================================================================
// Final correlation: sum_l ( tiles2[l,:] . W_norm[l,:] ) * lat_mean[l].
// Window = lat_mean (meanBuf * 0.2), W_norm = raw * KK / rowsum.
// Last use of lat_weights -> non-temporal loads.
// Deterministic reduction: per-block partial -> serial finish kernel.
// =====================================================================
__global__ __launch_bounds__(256)
void cm_corr_kernel(const float* __restrict__ meanBuf,
                    const float* __restrict__ Wlat,
                    const float* __restrict__ envK,
                    float* __restrict__ partials)    // (128)
{
    const int j0   = blockIdx.x * 16;
    const int i0   = blockIdx.y * 8;
    const int tid  = threadIdx.x;
    const int wave = tid >> 5;
    const int lane = tid & 31;
    const int m    = lane & 15;
    const int hi   = lane >> 4;

    __shared__ float win[48 * 56];
    __shared__ float env[KK];
    __shared__ int   off[KK];
    __shared__ float part[8];

    for (int idx = tid; idx < 48 * 56; idx += 256) {
        int wy = idx / 56, wx = idx % 56;
        int gi = i0 + wy - 20, gj = j0 + wx - 20;
        float v = 0.f;
        if (gi >= 0 && gi < S && gj >= 0 && gj < S) v = meanBuf[gi * S + gj] * 0.2f;
        win[idx] = v;
    }
    for (int k = tid; k < KK; k += 256) {
        env[k] = envK[k];
        off[k] = (k / LK) * 56 + (k % LK);
    }
    __syncthreads();

    const int li = i0 + wave;
    const int lm = li * S + j0 + m;
    const float* __restrict__ Wrow = Wlat + (size_t)lm * KK;
    const int abase = wave * 56 + m;

    vf8 acc  = {0.f,0.f,0.f,0.f,0.f,0.f,0.f,0.f};
    vf8 accS = {0.f,0.f,0.f,0.f,0.f,0.f,0.f,0.f};
    const vf2 ones = {1.f, 1.f};

    const int KMAIN = (KK / 4) * 4;
    for (int cb = 0; cb < KMAIN; cb += 4) {
        __builtin_prefetch(Wrow + cb + 96, 0, 0);
        int ka = cb + 2 * hi;
        vf2 a = {win[abase + off[ka]]     * env[ka],
                 win[abase + off[ka + 1]] * env[ka + 1]};
        vf2 b = {__builtin_nontemporal_load(Wrow + cb + hi),
                 __builtin_nontemporal_load(Wrow + cb + 2 + hi)};
        acc  = __builtin_amdgcn_wmma_f32_16x16x4_f32(false, a,    false, b, (short)0, acc,  false, false);
        accS = __builtin_amdgcn_wmma_f32_16x16x4_f32(false, ones, false, b, (short)0, accS, false, false);
    }
    {
        int cb = KMAIN;
        int ka = cb + 2 * hi;
        vf2 a = {0.f, 0.f};
        vf2 b = {0.f, 0.f};
        if (ka < KK)     a.x = win[abase + off[ka]]     * env[ka];
        if (ka + 1 < KK) a.y = win[abase + off[ka + 1]] * env[ka + 1];
        if (cb + hi     < KK) b.x = __builtin_nontemporal_load(Wrow + cb + hi);
        if (cb + 2 + hi < KK) b.y = __builtin_nontemporal_load(Wrow + cb + 2 + hi);
        acc  = __builtin_amdgcn_wmma_f32_16x16x4_f32(false, a,    false, b, (short)0, acc,  false, false);
        accS = __builtin_amdgcn_wmma_f32_16x16x4_f32(false, ones, false, b, (short)0, accS, false, false);
    }

    float dotv = 0.f, sumv = 0.f;
#pragma unroll
    for (int r = 0; r < 8; ++r) {
        float dlo = __shfl(acc[r],  r,      32);
        float dhi = __shfl(acc[r],  r + 24, 32);
        float slo = __shfl(accS[r], r,      32);
        float shi = __shfl(accS[r], r + 24, 32);
        if (lane == r)     { dotv = dlo; sumv = slo; }
        if (lane == r + 8) { dotv = dhi; sumv = shi; }
    }
    float contrib = 0.f;
    if (lane < 16) {
        int l = li * S + j0 + lane;
        contrib = dotv * ((float)KK / sumv) * (meanBuf[l] * 0.2f);
    }
#pragma unroll
    for (int o = 16; o > 0; o >>= 1) contrib += __shfl_down(contrib, o, 32);
    if (lane == 0) part[wave] = contrib;
    __syncthreads();
    if (tid == 0) {
        float t = 0.f;
        for (int w = 0; w < 8; ++w) t += part[w];
        partials[blockIdx.y * gridDim.x + blockIdx.x] = t;
    }
}

__global__ void cm_corr_finish(const float* __restrict__ partials, float* __restrict__ corr_out)
{
    if (threadIdx.x == 0 && blockIdx.x == 0) {
        float t = 0.f;
        for (int i = 0; i < 128; ++i) t += partials[i];
        corr_out[0] = t;
    }
}

// =====================================================================
extern "C" void kernel_launch(void* const* d_in, const int* in_sizes, int n_in,
                              void* d_out, int out_size, void* d_ws, size_t ws_size,
                              hipStream_t stream)
{
    (void)in_sizes; (void)n_in; (void)out_size; (void)ws_size;
    const float* x     = (const float*)d_in[0];   // (1,AC,HH,HH)
    const float* rfs   = (const float*)d_in[1];   // (S*S, PP, 1)
    const float* wlat  = (const float*)d_in[2];   // (S*S, KK, 1)
    const float* adath = (const float*)d_in[3];   // (1,1,S,S)
    const float* envA  = (const float*)d_in[4];   // (1,PP,1)
    const float* sre   = (const float*)d_in[5];   // (1,1,EX,EX)
    const float* envK  = (const float*)d_in[6];   // (1,KK,1)
    const float* mmask = (const float*)d_in[7];   // (1,361,1)

    float* out      = (float*)d_out;
    float* raw_out  = out;                 // 16384
    float* lat_out  = out + S * S;         // 16384
    float* corr_out = out + 2 * S * S;     // 1
    float* xt_out   = out + 2 * S * S + 1; // S*S*PP

    float* ws    = (float*)d_ws;
    float* aff   = ws;                  // 16384
    float* bufA  = ws + 1 * S * S;      // current lat
    float* bufB  = ws + 2 * S * S;      // conv result
    float* bufC  = ws + 3 * S * S;      // post-lateral
    float* meanB = ws + 4 * S * S;      // running sum of normalized lat
    float* parts = ws + 5 * S * S;      // 128 block partials

    dim3 blk(256);
    dim3 gridL(8, 16);   // 16x8 pixel tiles (WMMA kernels)
    dim3 gridS(8, 8);    // 16x16 pixel tiles

    cm_aff_kernel<<<gridL, blk, 0, stream>>>(x, rfs, envA, adath, raw_out, aff, bufA, xt_out);

    for (int it = 0; it < NITERS; ++it) {
        cm_sre_kernel<<<gridS, blk, 0, stream>>>(bufA, sre, bufB);
        cm_lat_kernel<<<gridL, blk, 0, stream>>>(bufB, wlat, envK, aff, bufC);
        cm_max_kernel<<<gridS, blk, 0, stream>>>(bufC, mmask, bufA, meanB, lat_out, it == 0 ? 1 : 0);
    }

    cm_corr_kernel<<<gridL, blk, 0, stream>>>(meanB, wlat, envK, parts);
    cm_corr_finish<<<dim3(1), dim3(32), 0, stream>>>(parts, corr_out);
}